// CRF_4337916969163
// MI455X (gfx1250) — compile-verified
//
#include <hip/hip_runtime.h>
#include <cstdint>

// Problem constants (from reference)
#define NTAGS     34
#define BB        1024
#define TT        1024
#define START_TAG 32
#define END_TAG   33
#define NEGINF    (-6969.0f)

// Forward pass: TDM feature chunking (timesteps per chunk)
#define CH     64
#define NCHUNK (TT / CH)
// Backtrack: backpointer rows staged per chunk
#define CT  256
#define NBT (TT / CT)

// Workspace layout in d_ws:
//   [0, BB*TT*NTAGS)                : backpointers, 1 byte each   (~35.65 MB)
//   [BB*TT*NTAGS, +BB*4)            : final argmax index per b (int32)
#define BP_BYTES ((size_t)BB * TT * NTAGS)

typedef unsigned int u32x4 __attribute__((ext_vector_type(4)));
typedef int          i32x8 __attribute__((ext_vector_type(8)));
typedef int          i32x4 __attribute__((ext_vector_type(4)));

#if defined(__HIP_DEVICE_COMPILE__) && __has_builtin(__builtin_amdgcn_tensor_load_to_lds)
#define HAVE_TDM 1
#else
#define HAVE_TDM 0
#endif

#if HAVE_TDM
// Issue one Tensor-DMA load of a [CH x NTAGS] f32 tile (row stride = NTAGS)
// from global memory into LDS. Wave-level op, tracked by TENSORcnt.
// This toolchain exposes the 6-arg builtin:
//   (uint32x4 g0, int32x8 g1, int32x4 g2, int32x4 g3, int32x8 extra, i32 cpol)
__device__ __forceinline__ void tdm_load_chunk(const float* gsrc, uint32_t lds_byte_addr) {
  uint64_t ga = (uint64_t)(uintptr_t)gsrc;
  u32x4 g0;
  g0.x = 1u;                                                 // count=1, user descriptor
  g0.y = lds_byte_addr;                                      // LDS dest (bytes)
  g0.z = (uint32_t)ga;                                       // global_addr[31:0]
  g0.w = ((uint32_t)(ga >> 32) & 0x01FFFFFFu) | 0x80000000u; // addr[56:32] | type=2
  i32x8 g1;
  g1[0] = (int)(2u << 16);                 // workgroup_mask=0 (no cluster), data_size=4B
  g1[1] = (int)((uint32_t)NTAGS << 16);    // tensor_dim0 = 34 (lo16 in hi half)
  g1[2] = (int)((uint32_t)CH << 16);       // tensor_dim0 hi=0 | tensor_dim1 = CH
  g1[3] = (int)((uint32_t)NTAGS << 16);    // tensor_dim1 hi=0 | tile_dim0 = 34
  g1[4] = (int)CH;                         // tile_dim1 = CH | tile_dim2 = 0
  g1[5] = NTAGS;                           // tensor_dim0_stride = 34 elements
  g1[6] = 0;                               // stride hi | tensor_dim1_stride lo (unused, 2D)
  g1[7] = 0;
  i32x4 gz4 = {0, 0, 0, 0};                // groups 2/3: dims/tiles unused (2D tile)
  i32x8 gz8 = {0, 0, 0, 0, 0, 0, 0, 0};    // extra operand (clang-23 form): zero
  __builtin_amdgcn_tensor_load_to_lds(g0, g1, gz4, gz4, gz8, 0);
}
#endif

__device__ __forceinline__ float bcast_lane(float v, int lane) {
  return __builtin_bit_cast(float,
      __builtin_amdgcn_readlane(__builtin_bit_cast(int, v), lane));
}

// ---------------------------------------------------------------------------
// Kernel 1: forward Viterbi recurrence. One wave (32 lanes) per batch element.
// lane L owns state j=L; lanes 0/1 additionally own states 32/33.
// Scores are wave-uniform (SGPRs via v_readlane); transition rows per-lane VGPRs.
// Features stream in via TDM double-buffered LDS tiles (s_wait_tensorcnt fenced).
// ---------------------------------------------------------------------------
__global__ __launch_bounds__(32) void viterbi_forward(
    const float* __restrict__ feat,   // [BB, TT, NTAGS]
    const float* __restrict__ trans,  // [NTAGS, NTAGS]
    float* __restrict__ out,          // [BB] best, then [BB, TT+1] path
    uint8_t* __restrict__ bp,         // [BB, TT, NTAGS] backpointers
    int* __restrict__ wsIdx)          // [BB] final argmax
{
  const int lane = threadIdx.x;
  const int b    = blockIdx.x;
  __shared__ float fbuf[2][CH * NTAGS];

  // Per-lane transition rows: trow = trans[j=lane][*], trow2 = trans[32+(lane&1)][*]
  const int j2 = START_TAG + (lane & 1);
  float trow[NTAGS], trow2[NTAGS];
#pragma unroll
  for (int k = 0; k < NTAGS; ++k) {
    trow[k]  = trans[lane * NTAGS + k];
    trow2[k] = trans[j2 * NTAGS + k];
  }
  const float tEndA = trans[END_TAG * NTAGS + lane]; // trans[END][j=lane]
  const float tEndB = trans[END_TAG * NTAGS + j2];   // trans[END][j2]

  // Wave-uniform score vector (held in SGPRs after readlane broadcasts)
  float sc[NTAGS];
#pragma unroll
  for (int k = 0; k < NTAGS; ++k) sc[k] = (k == START_TAG) ? 0.0f : NEGINF;

  const float* fb = feat + (size_t)b * TT * NTAGS;
  uint8_t* bpRow  = bp + (size_t)b * TT * NTAGS + lane;

  float m = NEGINF, m2 = NEGINF;

#if HAVE_TDM
  tdm_load_chunk(fb, (uint32_t)(uintptr_t)&fbuf[0][0]);
#endif

  for (int c = 0; c < NCHUNK; ++c) {
    const int cur = c & 1;
#if HAVE_TDM
    if (c + 1 < NCHUNK) {
      tdm_load_chunk(fb + (size_t)(c + 1) * CH * NTAGS,
                     (uint32_t)(uintptr_t)&fbuf[cur ^ 1][0]);
      __builtin_amdgcn_s_wait_tensorcnt(1);  // chunk c resident, c+1 in flight
    } else {
      __builtin_amdgcn_s_wait_tensorcnt(0);
    }
#else
    {
      const float* src = fb + (size_t)c * CH * NTAGS;
      for (int i = lane; i < CH * NTAGS; i += 32) fbuf[cur][i] = src[i];
    }
#endif
    volatile const float* fl = &fbuf[cur][0];

    for (int ti = 0; ti < CH; ++ti) {
      const float f  = fl[ti * NTAGS + lane];                      // feat[b,t,lane]
      const float f2 = fl[ti * NTAGS + START_TAG + (lane & 1)];    // feat[b,t,j2]

      m = NEGINF; m2 = NEGINF;
      int mi = 0, mi2 = 0;
#pragma unroll
      for (int k = 0; k < NTAGS; ++k) {
        const float s  = sc[k];           // uniform (SGPR) operand
        const float v  = s + trow[k];
        const float v2 = s + trow2[k];
        if (v  > m)  { m  = v;  mi  = k; }  // strict '>' keeps first max (jnp.argmax)
        if (v2 > m2) { m2 = v2; mi2 = k; }
      }
      m  += f;
      m2 += f2;

      // Broadcast new scores to all lanes as uniform values.
#pragma unroll
      for (int k = 0; k < 32; ++k) sc[k] = bcast_lane(m, k);
      sc[START_TAG] = bcast_lane(m2, 0);
      sc[END_TAG]   = bcast_lane(m2, 1);

      // Backpointers: 34 contiguous bytes per (b,t).
      const size_t off = (size_t)(c * CH + ti) * NTAGS;
      bpRow[off] = (uint8_t)mi;
      if (lane < 2) bpRow[off + START_TAG] = (uint8_t)mi2;  // states 32/33
    }
  }

  // Finalize: scores += trans[END]; wave-wide argmax with first-max tie-break.
  float bv = m + tEndA;
  int   bj = lane;
#pragma unroll
  for (int off = 16; off >= 1; off >>= 1) {
    const float ov = __shfl_xor(bv, off, 32);
    const int   oj = __shfl_xor(bj, off, 32);
    if (ov > bv || (ov == bv && oj < bj)) { bv = ov; bj = oj; }
  }
  const float v2f = m2 + tEndB;
  const float c32 = bcast_lane(v2f, 0);
  const float c33 = bcast_lane(v2f, 1);
  if (c32 > bv) { bv = c32; bj = START_TAG; }
  if (c33 > bv) { bv = c33; bj = END_TAG; }

  if (lane == 0) {
    out[b] = bv;                                        // best score
    out[BB + (size_t)b * (TT + 1) + TT] = (float)bj;    // last path entry = argmax
    wsIdx[b] = bj;
  }
}

// ---------------------------------------------------------------------------
// Kernel 2: backtrack. One wave per batch element. The pointer chase is serial
// in t, but the backpointer rows are not data-dependent: cooperatively stage
// CT rows (CT*34 bytes) into LDS, then lane 0 chases through LDS.
// ---------------------------------------------------------------------------
__global__ __launch_bounds__(32) void viterbi_backtrack(
    const uint8_t* __restrict__ bp,
    const int* __restrict__ wsIdx,
    float* __restrict__ out)
{
  const int lane = threadIdx.x;
  const int b    = blockIdx.x;
  __shared__ uint32_t lbuf[CT * NTAGS / 4];

  const uint32_t* src0 = (const uint32_t*)(bp + (size_t)b * TT * NTAGS);
  float* po = out + BB + (size_t)b * (TT + 1);
  int idx = wsIdx[b];

  for (int c = NBT - 1; c >= 0; --c) {
    const uint32_t* src = src0 + (size_t)c * (CT * NTAGS / 4);
    for (int i = lane; i < CT * NTAGS / 4; i += 32) lbuf[i] = src[i];
    __syncthreads();
    if (lane == 0) {
      volatile const uint8_t* lb = (volatile const uint8_t*)lbuf;
      for (int t = CT - 1; t >= 0; --t) {
        idx = lb[t * NTAGS + idx];
        po[c * CT + t] = (float)idx;
      }
    }
    __syncthreads();
    idx = __shfl(idx, 0, 32);  // carry chase state across chunks
  }
}

extern "C" void kernel_launch(void* const* d_in, const int* in_sizes, int n_in,
                              void* d_out, int out_size, void* d_ws, size_t ws_size,
                              hipStream_t stream) {
  const float* feat  = (const float*)d_in[0];  // [1024,1024,34] f32
  const float* trans = (const float*)d_in[1];  // [34,34] f32
  float* out = (float*)d_out;                  // 1024 + 1024*1025 elements

  uint8_t* bp    = (uint8_t*)d_ws;                      // 35.65 MB backpointers
  int*     wsIdx = (int*)((uint8_t*)d_ws + BP_BYTES);   // 4 KB final indices

  viterbi_forward  <<<BB, 32, 0, stream>>>(feat, trans, out, bp, wsIdx);
  viterbi_backtrack<<<BB, 32, 0, stream>>>(bp, wsIdx, out);
}